// GATGraphClassifier_24773371363343
// MI455X (gfx1250) — compile-verified
//
#include <hip/hip_runtime.h>
#include <math.h>

typedef __attribute__((ext_vector_type(2))) float v2f;
typedef __attribute__((ext_vector_type(8))) float v8f;

#define D 64
#define LDSW 66  // padded row stride (floats) to avoid 16-way LDS bank conflicts

// order-preserving float -> uint encoding for atomic max over floats
__device__ __forceinline__ unsigned encf(float f) {
  unsigned u = __float_as_uint(f);
  return (u & 0x80000000u) ? ~u : (u | 0x80000000u);
}
__device__ __forceinline__ float decf(unsigned u) {
  unsigned w = (u & 0x80000000u) ? (u & 0x7fffffffu) : ~u;
  return __uint_as_float(w);
}

// ---------------------------------------------------------------------------
// q = x@Wq^T + bq ; k = x@(Wk*mask)^T ; v = x@Wv^T + bv  via WMMA f32 16x16x4
// 128 threads = 4 waves; each wave owns a 16-node tile.
// A(m,k): lane = m + 16*(k>>1), vgpr = k&1. D(m,n): lane = n + 16*(m>=8), vgpr = m&7.
// K-loop is OUTER with 4 independent accumulators (one per 16-wide n-tile) so
// consecutive WMMAs never have a C->C serial dependency (no s_delay_alu stalls).
// ---------------------------------------------------------------------------
__global__ __launch_bounds__(128)
void proj_kernel(const float* __restrict__ x,
                 const float* __restrict__ Wq, const float* __restrict__ bq,
                 const float* __restrict__ Wk,
                 const float* __restrict__ Wv, const float* __restrict__ bv,
                 float* __restrict__ q, float* __restrict__ k, float* __restrict__ v,
                 int N)
{
  __shared__ float lw[3 * D * LDSW];
  const int tid = threadIdx.x;

  // Stage weights into LDS; zero input column 0 of W_key (the mask).
  for (int idx = tid; idx < 3 * D * D; idx += 128) {
    int p = idx >> 12;          // which matrix (D*D == 4096)
    int r = (idx >> 6) & 63;    // output row n
    int c = idx & 63;           // input col  k
    const float* W = (p == 0) ? Wq : (p == 1) ? Wk : Wv;
    float val = W[r * D + c];
    if (p == 1 && c == 0) val = 0.0f;
    lw[(p * D + r) * LDSW + c] = val;
  }
  __syncthreads();

  const int wave  = tid >> 5;
  const int lane  = tid & 31;
  const int node0 = (blockIdx.x * 4 + wave) * 16;
  if (node0 >= N) return;   // no barriers after this point

  const int m     = lane & 15;
  const int khalf = (lane >> 4) * 2;
  const int row   = min(node0 + m, N - 1);

  // Preload this lane's half of the x row: k = 4*kk + khalf + {0,1}
  v2f xa[16];
#pragma unroll
  for (int kk = 0; kk < 16; ++kk) {
    xa[kk] = *(const v2f*)(x + (size_t)row * D + 4 * kk + khalf);
  }

  const int n = lane & 15;     // output column within 16-wide tile
#pragma unroll
  for (int p = 0; p < 3; ++p) {
    const float* lwp  = &lw[p * D * LDSW];
    float*       dst  = (p == 0) ? q : (p == 1) ? k : v;
    const float* bias = (p == 0) ? bq : (p == 2) ? bv : nullptr;

    v8f acc[4];
#pragma unroll
    for (int nt = 0; nt < 4; ++nt)
      acc[nt] = (v8f){0.f, 0.f, 0.f, 0.f, 0.f, 0.f, 0.f, 0.f};

#pragma unroll
    for (int kk = 0; kk < 16; ++kk) {
      const v2f a = xa[kk];
#pragma unroll
      for (int nt = 0; nt < 4; ++nt) {
        // B(k,n) = W[n][k]  (q = x @ W^T); same lane/k layout as A
        v2f b = *(const v2f*)&lwp[(nt * 16 + n) * LDSW + 4 * kk + khalf];
        acc[nt] = __builtin_amdgcn_wmma_f32_16x16x4_f32(false, a, false, b,
                                                        (short)0, acc[nt],
                                                        false, false);
      }
    }

#pragma unroll
    for (int nt = 0; nt < 4; ++nt) {
      const float badd = bias ? bias[nt * 16 + n] : 0.0f;
#pragma unroll
      for (int i = 0; i < 8; ++i) {
        int mr    = i + 8 * (lane >> 4);
        int nodeR = node0 + mr;
        if (nodeR < N) dst[(size_t)nodeR * D + nt * 16 + n] = acc[nt][i] + badd;
      }
    }
  }
}

// ---------------------------------------------------------------------------
// Per-edge scores: s = dot(q[dst], k[src]) / 8 ; track global max (encoded).
// One wave per edge; edges E..E+N-1 are implicit self-loops.
// ---------------------------------------------------------------------------
__global__ __launch_bounds__(256)
void score_kernel(const float* __restrict__ q, const float* __restrict__ k,
                  const int* __restrict__ ei, int E, int N,
                  float* __restrict__ scores, unsigned* __restrict__ maxenc)
{
  const int lane = threadIdx.x & 31;
  const int wpb  = blockDim.x >> 5;
  const int gw   = blockIdx.x * wpb + (threadIdx.x >> 5);
  const int nw   = gridDim.x * wpb;
  const int M    = E + N;
  float wmax = -3.4e38f;
  for (int e = gw; e < M; e += nw) {
    int s, d;
    if (e < E) { s = ei[e]; d = ei[E + e]; } else { s = d = e - E; }
    v2f qv = *(const v2f*)(q + (size_t)d * D + 2 * lane);
    v2f kv = *(const v2f*)(k + (size_t)s * D + 2 * lane);
    float p = qv.x * kv.x + qv.y * kv.y;
#pragma unroll
    for (int off = 16; off > 0; off >>= 1) p += __shfl_xor(p, off, 32);
    p *= 0.125f;   // 1/sqrt(64)
    if (lane == 0) scores[e] = p;
    wmax = fmaxf(wmax, p);
  }
  if (lane == 0) atomicMax(maxenc, encf(wmax));
}

// ---------------------------------------------------------------------------
// alpha[e] = exp(s - max) in place; accumulate global sum.
// ---------------------------------------------------------------------------
__global__ __launch_bounds__(256)
void expsum_kernel(float* __restrict__ scores, int M,
                   const unsigned* __restrict__ maxenc, float* __restrict__ sumexp)
{
  const float mx = decf(*maxenc);
  float local = 0.f;
  const int i0     = blockIdx.x * blockDim.x + threadIdx.x;
  const int stride = gridDim.x * blockDim.x;
  for (int i = i0; i < M; i += stride) {
    float e = expf(scores[i] - mx);
    scores[i] = e;
    local += e;
  }
#pragma unroll
  for (int off = 16; off > 0; off >>= 1) local += __shfl_xor(local, off, 32);
  if ((threadIdx.x & 31) == 0) atomicAdd(sumexp, local);
}

// ---------------------------------------------------------------------------
// out[dst] += v[src] * alpha[e] / sum   (f32 global atomics, wave per edge)
// ---------------------------------------------------------------------------
__global__ __launch_bounds__(256)
void message_kernel(const float* __restrict__ v, const float* __restrict__ alpha,
                    const int* __restrict__ ei, int E, int N,
                    const float* __restrict__ sumexp, float* __restrict__ out)
{
  const float inv = 1.0f / sumexp[0];
  const int lane = threadIdx.x & 31;
  const int wpb  = blockDim.x >> 5;
  const int gw   = blockIdx.x * wpb + (threadIdx.x >> 5);
  const int nw   = gridDim.x * wpb;
  const int M    = E + N;
  for (int e = gw; e < M; e += nw) {
    int s, d;
    if (e < E) { s = ei[e]; d = ei[E + e]; } else { s = d = e - E; }
    const float a = alpha[e] * inv;
    v2f vv = *(const v2f*)(v + (size_t)s * D + 2 * lane);
    atomicAdd(out + (size_t)d * D + 2 * lane,     vv.x * a);
    atomicAdd(out + (size_t)d * D + 2 * lane + 1, vv.y * a);
  }
}

// ---------------------------------------------------------------------------
// h = relu(out); per-graph sums + counts (wave per node)
// ---------------------------------------------------------------------------
__global__ __launch_bounds__(256)
void pool_kernel(const float* __restrict__ out, const int* __restrict__ batch,
                 int N, float* __restrict__ psum, float* __restrict__ pcnt)
{
  const int lane = threadIdx.x & 31;
  const int wpb  = blockDim.x >> 5;
  const int gw   = blockIdx.x * wpb + (threadIdx.x >> 5);
  const int nw   = gridDim.x * wpb;
  for (int n = gw; n < N; n += nw) {
    const int g = batch[n];
    v2f h = *(const v2f*)(out + (size_t)n * D + 2 * lane);
    atomicAdd(psum + g * D + 2 * lane,     fmaxf(h.x, 0.f));
    atomicAdd(psum + g * D + 2 * lane + 1, fmaxf(h.y, 0.f));
    if (lane == 0) atomicAdd(pcnt + g, 1.0f);
  }
}

// ---------------------------------------------------------------------------
// logits[g][c] = (psum[g]/max(cnt,1)) . W_fc[c] + b_fc[c]
// ---------------------------------------------------------------------------
__global__ __launch_bounds__(128)
void fc_kernel(const float* __restrict__ psum, const float* __restrict__ pcnt,
               const float* __restrict__ Wfc, const float* __restrict__ bfc,
               int G, int C, float* __restrict__ outp)
{
  const int t = blockIdx.x * blockDim.x + threadIdx.x;
  if (t >= G * C) return;
  const int g = t / C, c = t % C;
  const float inv = 1.0f / fmaxf(pcnt[g], 1.0f);
  float acc = bfc[c];
  for (int dd = 0; dd < D; ++dd)
    acc += (psum[g * D + dd] * inv) * Wfc[c * D + dd];
  outp[t] = acc;
}

// ---------------------------------------------------------------------------
extern "C" void kernel_launch(void* const* d_in, const int* in_sizes, int n_in,
                              void* d_out, int out_size, void* d_ws, size_t ws_size,
                              hipStream_t stream)
{
  const float* x     = (const float*)d_in[0];
  const int*   ei    = (const int*)  d_in[1];
  const int*   batch = (const int*)  d_in[2];
  const float* Wq    = (const float*)d_in[3];
  const float* bq    = (const float*)d_in[4];
  const float* Wk    = (const float*)d_in[5];
  const float* Wv    = (const float*)d_in[6];
  const float* bv    = (const float*)d_in[7];
  const float* Wfc   = (const float*)d_in[8];
  const float* bfc   = (const float*)d_in[9];
  float* out = (float*)d_out;

  const int N = in_sizes[0] / D;
  const int E = in_sizes[1] / 2;
  const int C = in_sizes[8] / D;
  const int G = out_size / C;
  const int M = E + N;

  char* ws = (char*)d_ws;
  size_t off = 0;
  auto alloc = [&](size_t bytes) -> char* {
    char* p = ws + off;
    off = (off + bytes + 255) & ~(size_t)255;
    return p;
  };
  float*    q      = (float*)   alloc((size_t)N * D * 4);
  float*    k      = (float*)   alloc((size_t)N * D * 4);
  float*    v      = (float*)   alloc((size_t)N * D * 4);
  float*    h      = (float*)   alloc((size_t)N * D * 4);
  float*    sc     = (float*)   alloc((size_t)M * 4);
  float*    psum   = (float*)   alloc((size_t)G * D * 4);
  float*    pcnt   = (float*)   alloc((size_t)G * 4);
  unsigned* maxenc = (unsigned*)alloc(4);
  float*    sumexp = (float*)   alloc(4);
  (void)ws_size; (void)n_in;

  // Accumulation buffers must be re-zeroed every launch (graph-capturable).
  hipMemsetAsync(h,      0, (size_t)N * D * 4, stream);
  hipMemsetAsync(psum,   0, (size_t)G * D * 4, stream);
  hipMemsetAsync(pcnt,   0, (size_t)G * 4,     stream);
  hipMemsetAsync(maxenc, 0, 4,                 stream);
  hipMemsetAsync(sumexp, 0, 4,                 stream);

  proj_kernel   <<<(N + 63) / 64,        128, 0, stream>>>(x, Wq, bq, Wk, Wv, bv, q, k, v, N);
  score_kernel  <<<2048,                 256, 0, stream>>>(q, k, ei, E, N, sc, maxenc);
  expsum_kernel <<<2048,                 256, 0, stream>>>(sc, M, maxenc, sumexp);
  message_kernel<<<4096,                 256, 0, stream>>>(v, sc, ei, E, N, sumexp, h);
  pool_kernel   <<<2048,                 256, 0, stream>>>(h, batch, N, psum, pcnt);
  fc_kernel     <<<(G * C + 127) / 128,  128, 0, stream>>>(psum, pcnt, Wfc, bfc, G, C, out);
}